// DiffusionStep_64828236366586
// MI455X (gfx1250) — compile-verified
//
#include <hip/hip_runtime.h>
#include <stdint.h>

// -----------------------------------------------------------------------------
// DiffusionStep (Chebyshev K=4) for MI455X / gfx1250.
// propagate() = edge-parallel gather/scale/scatter-add SpMM, L2-resident
// (features 12.8MB + edges 15MB + acc 12.8MB << 192MB L2).
// Edge stream staged to LDS with the Tensor Data Mover (double buffered,
// s_wait_tensorcnt), gather rows prefetched with global_prefetch_b8,
// scatter via explicit non-returning global_atomic_add_f32 (inline asm,
// SADDR form, scope:SCOPE_DEV).
// -----------------------------------------------------------------------------

typedef uint32_t u32;
typedef uint64_t u64;
typedef __attribute__((ext_vector_type(4))) u32 v4u;
typedef __attribute__((ext_vector_type(8))) int v8i;
typedef __attribute__((ext_vector_type(4))) int v4i;

#define TILE 256   // edges per TDM tile
#define DFEAT 64   // feature dim

// 1-D TDM tile load: tile_elems 4-byte elements from gptr -> LDS offset lds_off.
// tensor_dim0 = valid_elems so tail reads past E return zero (OOB -> 0).
__device__ __forceinline__ void tdm_load_1d_b32(u32 lds_off, const void* gptr,
                                                u32 tile_elems, u32 valid_elems) {
    u64 ga = (u64)(uintptr_t)gptr;
    v4u g0;
    g0.x = 1u;                                            // count=1, user descriptor
    g0.y = lds_off;                                       // lds_addr
    g0.z = (u32)ga;                                       // global_addr[31:0]
    g0.w = (u32)((ga >> 32) & 0x01FFFFFFu) | 0x80000000u; // global_addr[56:32] | type=2
    v8i g1;
    g1[0] = (int)(2u << 16);                              // data_size=2 (4B), mask=0
    g1[1] = (int)((valid_elems & 0xFFFFu) << 16);         // tensor_dim0[15:0]
    g1[2] = (int)((valid_elems >> 16) | (1u << 16));      // tensor_dim0[31:16], tensor_dim1=1
    g1[3] = (int)((tile_elems & 0xFFFFu) << 16);          // tile_dim0
    g1[4] = 0;                                            // tile_dim1=0 (1-D), tile_dim2=0
    g1[5] = (int)valid_elems;                             // tensor_dim0_stride (unused, 1 row)
    g1[6] = 0;
    g1[7] = 0;
    v4i z4 = (v4i)0;
#if defined(__has_include)
#if __has_include(<hip/amd_detail/amd_gfx1250_TDM.h>)
    v8i z8 = (v8i)0;
    __builtin_amdgcn_tensor_load_to_lds(g0, g1, z4, z4, z8, 0);
#else
    __builtin_amdgcn_tensor_load_to_lds(g0, g1, z4, z4, 0);
#endif
#else
    __builtin_amdgcn_tensor_load_to_lds(g0, g1, z4, z4, 0);
#endif
}

// Non-returning device-scope f32 atomic add: base (uniform, SGPR pair) +
// per-lane byte offset + immediate offset. Tracked on STOREcnt; drained by
// the implicit wait-idle at s_endpgm.
__device__ __forceinline__ void atomic_add_f32_dev(float* base, u32 byteOff,
                                                   float v0, float v1,
                                                   float v2, float v3) {
    asm volatile("global_atomic_add_f32 %0, %1, %2 offset:0 scope:SCOPE_DEV"
                 :: "v"(byteOff), "v"(v0), "s"(base) : "memory");
    asm volatile("global_atomic_add_f32 %0, %1, %2 offset:4 scope:SCOPE_DEV"
                 :: "v"(byteOff), "v"(v1), "s"(base) : "memory");
    asm volatile("global_atomic_add_f32 %0, %1, %2 offset:8 scope:SCOPE_DEV"
                 :: "v"(byteOff), "v"(v2), "s"(base) : "memory");
    asm volatile("global_atomic_add_f32 %0, %1, %2 offset:12 scope:SCOPE_DEV"
                 :: "v"(byteOff), "v"(v3), "s"(base) : "memory");
}

// acc[dst] += w * h[src], edge-parallel. 16 lanes per edge (float4 per lane):
// one wave32 loads 2 contiguous 256B rows per issue; atomics per edge land in
// one contiguous 64-float row segment.
__global__ void __launch_bounds__(256)
diffusion_scatter(const float* __restrict__ h,
                  const int*  __restrict__ srcIdx,
                  const int*  __restrict__ dstIdx,
                  const float* __restrict__ ew,
                  float* __restrict__ acc,
                  int E, int nTiles)
{
    __shared__ int   s_src[2][TILE];
    __shared__ int   s_dst[2][TILE];
    __shared__ float s_w  [2][TILE];

    const int sub  = (int)(threadIdx.x >> 4);   // edge slot within sweep (0..15)
    const int lane = (int)(threadIdx.x & 15);   // float4 slot in the row
    const int fo   = lane << 2;                 // feature offset in floats

    int t   = (int)blockIdx.x;
    int buf = 0;

    // Prologue: wave 0 DMAs the first tile (TDM ignores EXEC; other waves skip).
    if (t < nTiles && threadIdx.x < 32) {
        const int base = t * TILE;
        const int rem  = E - base;
        const u32 valid = (u32)(rem < TILE ? rem : TILE);
        tdm_load_1d_b32((u32)(uintptr_t)&s_src[0][0], srcIdx + base, TILE, valid);
        tdm_load_1d_b32((u32)(uintptr_t)&s_dst[0][0], dstIdx + base, TILE, valid);
        tdm_load_1d_b32((u32)(uintptr_t)&s_w  [0][0], ew     + base, TILE, valid);
    }

    for (; t < nTiles; t += (int)gridDim.x) {
        const int tn = t + (int)gridDim.x;
        if (threadIdx.x < 32) {
            if (tn < nTiles) {
                // Issue next tile into the other buffer, then wait for the
                // current tile (3 oldest tensor ops) only: TENSORcnt <= 3.
                const int nbase = tn * TILE;
                const int nrem  = E - nbase;
                const u32 nvalid = (u32)(nrem < TILE ? nrem : TILE);
                const int nb = buf ^ 1;
                tdm_load_1d_b32((u32)(uintptr_t)&s_src[nb][0], srcIdx + nbase, TILE, nvalid);
                tdm_load_1d_b32((u32)(uintptr_t)&s_dst[nb][0], dstIdx + nbase, TILE, nvalid);
                tdm_load_1d_b32((u32)(uintptr_t)&s_w  [nb][0], ew     + nbase, TILE, nvalid);
                __builtin_amdgcn_s_wait_tensorcnt(3);
            } else {
                __builtin_amdgcn_s_wait_tensorcnt(0);
            }
        }
        __syncthreads();   // publish current tile's LDS to all 8 waves

        const int base = t * TILE;
        #pragma unroll 4
        for (int s = 0; s < TILE / 16; ++s) {
            const int j = (s << 4) + sub;
            // Prefetch gather rows ~3 sweeps ahead to hide L2 latency.
            const int jp = j + 48;
            if (jp < TILE && base + jp < E) {
                __builtin_prefetch(h + (size_t)s_src[buf][jp] * DFEAT + fo, 0, 3);
            }
            const int e = base + j;
            if (e < E) {
                const int   sn = s_src[buf][j];
                const int   dn = s_dst[buf][j];
                const float w  = s_w[buf][j];
                const float4 v = *reinterpret_cast<const float4*>(h + (size_t)sn * DFEAT + fo);
                const u32 byteOff = (((u32)dn * (u32)DFEAT) + (u32)fo) * 4u;
                atomic_add_f32_dev(acc, byteOff, v.x * w, v.y * w, v.z * w, v.w * w);
            }
        }
        __syncthreads();   // all waves done with buf before it is DMA'd again
        buf ^= 1;
    }
}

__global__ void __launch_bounds__(256)
diffusion_zero(float4* __restrict__ acc, int n4)
{
    const int i = (int)(blockIdx.x * blockDim.x + threadIdx.x);
    if (i < n4) {
        float4 z; z.x = 0.f; z.y = 0.f; z.z = 0.f; z.w = 0.f;
        acc[i] = z;
    }
}

// out0 = x ; out1 = x - acc ; acc = 0
__global__ void __launch_bounds__(256)
diffusion_combine_first(const float4* __restrict__ x, float4* __restrict__ acc,
                        float4* __restrict__ out0, float4* __restrict__ out1, int n4)
{
    const int i = (int)(blockIdx.x * blockDim.x + threadIdx.x);
    if (i < n4) {
        const float4 xv = x[i];
        const float4 av = acc[i];
        out0[i] = xv;
        float4 r;
        r.x = xv.x - av.x; r.y = xv.y - av.y; r.z = xv.z - av.z; r.w = xv.w - av.w;
        out1[i] = r;
        float4 z; z.x = 0.f; z.y = 0.f; z.z = 0.f; z.w = 0.f;
        acc[i] = z;
    }
}

// out = 2*(t1 - acc) - t0 ; acc = 0
__global__ void __launch_bounds__(256)
diffusion_combine_cheb(const float4* __restrict__ t1, const float4* __restrict__ t0,
                       float4* __restrict__ acc, float4* __restrict__ out, int n4)
{
    const int i = (int)(blockIdx.x * blockDim.x + threadIdx.x);
    if (i < n4) {
        const float4 a = t1[i];
        const float4 b = t0[i];
        const float4 c = acc[i];
        float4 r;
        r.x = 2.0f * (a.x - c.x) - b.x;
        r.y = 2.0f * (a.y - c.y) - b.y;
        r.z = 2.0f * (a.z - c.z) - b.z;
        r.w = 2.0f * (a.w - c.w) - b.w;
        out[i] = r;
        float4 z; z.x = 0.f; z.y = 0.f; z.z = 0.f; z.w = 0.f;
        acc[i] = z;
    }
}

extern "C" void kernel_launch(void* const* d_in, const int* in_sizes, int n_in,
                              void* d_out, int out_size, void* d_ws, size_t ws_size,
                              hipStream_t stream) {
    const float* x  = (const float*)d_in[0];   // [N, 64]
    const int*   ei = (const int*)d_in[1];     // [2, E]
    const float* ew = (const float*)d_in[2];   // [E]

    const int E  = in_sizes[2];
    const int ND = in_sizes[0];                // N * 64
    const size_t nd = (size_t)ND;

    const int* srcIdx = ei;
    const int* dstIdx = ei + E;

    float* out = (float*)d_out;                // [5, N, 64]
    float* acc = (float*)d_ws;                 // [N, 64] scratch accumulator

    const int n4 = ND / 4;
    const dim3 cb(256), cg((unsigned)((n4 + 255) / 256));

    const int nTiles = (E + TILE - 1) / TILE;
    const int sg = nTiles < 1024 ? nTiles : 1024;

    float* T0 = out;
    float* T1 = out + nd;
    float* T2 = out + 2 * nd;
    float* T3 = out + 3 * nd;
    float* T4 = out + 4 * nd;

    // acc must start at zero every call (d_ws is poisoned, not re-zeroed).
    diffusion_zero<<<cg, cb, 0, stream>>>((float4*)acc, n4);

    // A_x = A*x ; T0 = x ; T1 = x - A_x   (combine re-zeroes acc)
    diffusion_scatter<<<sg, 256, 0, stream>>>(x, srcIdx, dstIdx, ew, acc, E, nTiles);
    diffusion_combine_first<<<cg, cb, 0, stream>>>((const float4*)x, (float4*)acc,
                                                   (float4*)T0, (float4*)T1, n4);

    // T2 = 2*(T1 - A*T1) - T0
    diffusion_scatter<<<sg, 256, 0, stream>>>(T1, srcIdx, dstIdx, ew, acc, E, nTiles);
    diffusion_combine_cheb<<<cg, cb, 0, stream>>>((const float4*)T1, (const float4*)T0,
                                                  (float4*)acc, (float4*)T2, n4);

    // T3 = 2*(T2 - A*T2) - T1
    diffusion_scatter<<<sg, 256, 0, stream>>>(T2, srcIdx, dstIdx, ew, acc, E, nTiles);
    diffusion_combine_cheb<<<cg, cb, 0, stream>>>((const float4*)T2, (const float4*)T1,
                                                  (float4*)acc, (float4*)T3, n4);

    // T4 = 2*(T3 - A*T3) - T2
    diffusion_scatter<<<sg, 256, 0, stream>>>(T3, srcIdx, dstIdx, ew, acc, E, nTiles);
    diffusion_combine_cheb<<<cg, cb, 0, stream>>>((const float4*)T3, (const float4*)T2,
                                                  (float4*)acc, (float4*)T4, n4);
}